// GAE_58153857188526
// MI455X (gfx1250) — compile-verified
//
#include <hip/hip_runtime.h>

// ---------------------------------------------------------------------------
// GCN 3-layer pipeline for gfx1250 (MI455X).
// GEMMs use V_WMMA_F32_16X16X4_F32 (full fp32; GEMMs are bandwidth-trivial,
// the edge scatter dominates). Scatter uses hardware global_atomic_add_f32
// via __hip_atomic_fetch_add(relaxed, agent scope).
// Harness contract: integer inputs passed as const int*.
// ---------------------------------------------------------------------------

typedef __attribute__((ext_vector_type(2))) float v2f;
typedef __attribute__((ext_vector_type(8))) float v8f;

typedef int idx_t;   // harness maps integer inputs to const int*

__device__ __forceinline__ void atomic_add_f32(float* p, float v) {
  __hip_atomic_fetch_add(p, v, __ATOMIC_RELAXED, __HIP_MEMORY_SCOPE_AGENT);
}

// ---------------------------------------------------------------------------
// out[M x NC] = A[M x K] * W[K x NC]  (+bias)(+relu), fp32 WMMA 16x16x4.
// One wave per 16-row tile; flags are compile-time so the epilogue is
// branch-free. Optional fused self-loop init:
//   self_out[row][col] = dinv[row]^2 * out[row][col] (+ self_bias[col])
//
// VGPR layouts (ISA 7.12.2, 32-bit):
//   A 16x4: lanes 0-15 hold rows M=lid with VGPR0=K, VGPR1=K+1;
//           lanes 16-31 hold VGPR0=K+2, VGPR1=K+3.
//   B 4x16: VGPR0 lanes0-15 = row K (cols 0-15), lanes16-31 = row K+2;
//           VGPR1 lanes0-15 = row K+1, lanes16-31 = row K+3.
//   C/D:    VGPR r: lanes0-15 -> M=r, lanes16-31 -> M=8+r; N = lane%16.
// ---------------------------------------------------------------------------
template <int K, int NC, bool BIAS, bool RELU, bool SELFINIT, bool SELF_BIAS>
__global__ void __launch_bounds__(256)
wmma_gemm_kernel(const float* __restrict__ A, const float* __restrict__ W,
                 const float* __restrict__ bias, float* __restrict__ out,
                 const float* __restrict__ dinv, float* __restrict__ self_out,
                 const float* __restrict__ self_bias, int M) {
  const int lane = threadIdx.x & 31;
  const int wave = blockIdx.x * (blockDim.x >> 5) + (threadIdx.x >> 5);
  const int rowBase = wave * 16;
  if (rowBase + 16 > M) return;            // wave-uniform -> EXEC stays all-1s

  const int half = lane >> 4;              // 0: K pair {0,1}; 1: {2,3}
  const int lid  = lane & 15;

  const float* arow = A + (size_t)(rowBase + lid) * K;

  v8f acc[NC / 16];
#pragma unroll
  for (int t = 0; t < NC / 16; ++t) acc[t] = (v8f){0, 0, 0, 0, 0, 0, 0, 0};

#pragma unroll 4
  for (int k = 0; k < K; k += 4) {
    v2f a;
    a.x = arow[k + half * 2 + 0];
    a.y = arow[k + half * 2 + 1];
#pragma unroll
    for (int t = 0; t < NC / 16; ++t) {
      const float* wp = W + (size_t)(k + half * 2) * NC + t * 16 + lid;
      v2f b;
      b.x = wp[0];
      b.y = wp[NC];
      acc[t] = __builtin_amdgcn_wmma_f32_16x16x4_f32(
          false, a, false, b, (short)0, acc[t], false, false);
    }
  }

  // Branch-free epilogue: bias preloaded once per (lane, tile).
  float bval[NC / 16];
  float sbval[NC / 16];
  if constexpr (BIAS) {
#pragma unroll
    for (int t = 0; t < NC / 16; ++t) bval[t] = bias[t * 16 + lid];
  }
  if constexpr (SELF_BIAS) {
#pragma unroll
    for (int t = 0; t < NC / 16; ++t) sbval[t] = self_bias[t * 16 + lid];
  }
#pragma unroll
  for (int r = 0; r < 8; ++r) {
    const int row = rowBase + half * 8 + r;
    float di2 = 0.0f;
    if constexpr (SELFINIT) {
      const float di = dinv[row];
      di2 = di * di;
    }
#pragma unroll
    for (int t = 0; t < NC / 16; ++t) {
      const int col = t * 16 + lid;
      float v = acc[t][r];
      if constexpr (BIAS) v += bval[t];
      if constexpr (RELU) v = fmaxf(v, 0.0f);
      out[(size_t)row * NC + col] = v;
      if constexpr (SELFINIT) {
        float sv = di2 * v;
        if constexpr (SELF_BIAS) sv += sbval[t];
        self_out[(size_t)row * NC + col] = sv;
      }
    }
  }
}

// Scalar tail for M % 16 rows (never launched for this shape; robustness only).
template <int K, int NC, bool BIAS, bool RELU, bool SELFINIT, bool SELF_BIAS>
__global__ void gemm_tail_kernel(const float* __restrict__ A,
                                 const float* __restrict__ W,
                                 const float* __restrict__ bias,
                                 float* __restrict__ out,
                                 const float* __restrict__ dinv,
                                 float* __restrict__ self_out,
                                 const float* __restrict__ self_bias,
                                 int row0, int M) {
  int gid = blockIdx.x * blockDim.x + threadIdx.x;
  int row = row0 + gid / NC;
  int col = gid % NC;
  if (row >= M) return;
  float acc = 0.0f;
  const float* arow = A + (size_t)row * K;
#pragma unroll 8
  for (int k = 0; k < K; ++k) acc += arow[k] * W[(size_t)k * NC + col];
  if constexpr (BIAS) acc += bias[col];
  if constexpr (RELU) acc = fmaxf(acc, 0.0f);
  out[(size_t)row * NC + col] = acc;
  if constexpr (SELFINIT) {
    const float di = dinv[row];
    float sv = di * di * acc;
    if constexpr (SELF_BIAS) sv += self_bias[col];
    self_out[(size_t)row * NC + col] = sv;
  }
}

// ---------------------------------------------------------------------------
// Elementwise / graph kernels
// ---------------------------------------------------------------------------
__global__ void fill_kernel(float* __restrict__ p, float v, int n) {
  int i = blockIdx.x * blockDim.x + threadIdx.x;
  if (i < n) p[i] = v;
}

__global__ void deg_edge_kernel(const idx_t* __restrict__ dst,
                                const float* __restrict__ ew,
                                float* __restrict__ deg, int E) {
  int e = blockIdx.x * blockDim.x + threadIdx.x;
  if (e < E) atomic_add_f32(&deg[dst[e]], ew[e]);
}

__global__ void rsqrt_kernel(float* __restrict__ d, int n) {
  int i = blockIdx.x * blockDim.x + threadIdx.x;
  if (i < n) {
    float v = d[i];
    d[i] = (v > 0.0f) ? rsqrtf(v) : 0.0f;
  }
}

__global__ void norm_kernel(const idx_t* __restrict__ src,
                            const idx_t* __restrict__ dst,
                            const float* __restrict__ ew,
                            const float* __restrict__ dinv,
                            float* __restrict__ norm, int E) {
  int e = blockIdx.x * blockDim.x + threadIdx.x;
  if (e < E) norm[e] = dinv[src[e]] * dinv[dst[e]] * ew[e];
}

__global__ void bias_relu_kernel(float* __restrict__ h,
                                 const float* __restrict__ b, int total, int C) {
  int i = blockIdx.x * blockDim.x + threadIdx.x;
  if (i < total) h[i] = fmaxf(h[i] + b[i % C], 0.0f);
}

// agg[dst][g*4 .. g*4+3] += norm[e] * msg[src][g*4 .. g*4+3]
// Consecutive threads cover one edge's channel groups -> coalesced gathers,
// contiguous atomics. msg table (12.8 MB) stays L2-resident on 192 MB L2.
template <int C>
__global__ void __launch_bounds__(256)
scatter_kernel(const idx_t* __restrict__ src, const idx_t* __restrict__ dst,
               const float* __restrict__ norm, const float* __restrict__ msg,
               float* __restrict__ agg, int E) {
  constexpr int G = C / 4;
  constexpr int SH = (C == 32) ? 3 : 2;
  long long gid = (long long)blockIdx.x * blockDim.x + threadIdx.x;
  if (gid >= (long long)E * G) return;
  int e = (int)(gid >> SH);
  int g = (int)(gid & (G - 1));
  int s = src[e], d = dst[e];
  float w = norm[e];
  const float4 m = *(const float4*)(msg + (size_t)s * C + g * 4);
  float* ap = agg + (size_t)d * C + g * 4;
  atomic_add_f32(ap + 0, w * m.x);
  atomic_add_f32(ap + 1, w * m.y);
  atomic_add_f32(ap + 2, w * m.z);
  atomic_add_f32(ap + 3, w * m.w);
}

// ---------------------------------------------------------------------------
static inline int cdiv(long long a, long long b) { return (int)((a + b - 1) / b); }

extern "C" void kernel_launch(void* const* d_in, const int* in_sizes, int n_in,
                              void* d_out, int out_size, void* d_ws, size_t ws_size,
                              hipStream_t stream) {
  const float* x     = (const float*)d_in[0];   // [N,128]
  const idx_t* ei    = (const idx_t*)d_in[1];   // [2,E]
  const float* ew    = (const float*)d_in[2];   // [E]
  const float* lin_w = (const float*)d_in[3];   // [128,32]
  const float* lin_b = (const float*)d_in[4];   // [32]
  const float* c1_w  = (const float*)d_in[5];   // [32,32]
  const float* c1_b  = (const float*)d_in[6];   // [32]
  const float* c2_w  = (const float*)d_in[7];   // [32,16]
  const float* c2_b  = (const float*)d_in[8];   // [16]

  const int N = in_sizes[0] / 128;
  const int E = in_sizes[2];
  const idx_t* src = ei;
  const idx_t* dst = ei + E;

  // Workspace carve (~52 MB): dinv[N] | norm[E] | h0[N*32] | tmp[N*32] | agg[N*32]
  float* dinv = (float*)d_ws;
  float* norm = dinv + N;
  float* h0   = norm + E;
  float* tmp  = h0 + (size_t)N * 32;
  float* agg  = tmp + (size_t)N * 32;
  float* out  = (float*)d_out;

  const int fullWaves = N / 16;     // N = 100000 = 6250 * 16 (tail unused here)
  const int tail = N - fullWaves * 16;
  const int gemmBlocks = cdiv(fullWaves, 8);   // 8 waves (256 thr) / block

  // 1) h0 = relu(x @ lin_w + lin_b)
  wmma_gemm_kernel<128, 32, true, true, false, false>
      <<<gemmBlocks, 256, 0, stream>>>(x, lin_w, lin_b, h0, nullptr, nullptr, nullptr, N);
  if (tail)
    gemm_tail_kernel<128, 32, true, true, false, false>
        <<<cdiv((long long)tail * 32, 256), 256, 0, stream>>>(
            x, lin_w, lin_b, h0, nullptr, nullptr, nullptr, fullWaves * 16, N);

  // 2-4) deg (self-loop=1) -> atomically add edge weights -> dinv = rsqrt(deg)
  fill_kernel<<<cdiv(N, 256), 256, 0, stream>>>(dinv, 1.0f, N);
  deg_edge_kernel<<<cdiv(E, 256), 256, 0, stream>>>(dst, ew, dinv, E);
  rsqrt_kernel<<<cdiv(N, 256), 256, 0, stream>>>(dinv, N);

  // 5) norm[e] = dinv[src]*dinv[dst]*ew
  norm_kernel<<<cdiv(E, 256), 256, 0, stream>>>(src, dst, ew, dinv, norm, E);

  // ---- GCNConv 1 ----
  // 6+7) tmp = h0 @ c1_w ; fused: agg = dinv^2 * tmp  (self loop)
  wmma_gemm_kernel<32, 32, false, false, true, false>
      <<<gemmBlocks, 256, 0, stream>>>(h0, c1_w, nullptr, tmp, dinv, agg, nullptr, N);
  if (tail)
    gemm_tail_kernel<32, 32, false, false, true, false>
        <<<cdiv((long long)tail * 32, 256), 256, 0, stream>>>(
            h0, c1_w, nullptr, tmp, dinv, agg, nullptr, fullWaves * 16, N);
  // 8) agg[dst] += norm * tmp[src]
  scatter_kernel<32><<<cdiv((long long)E * 8, 256), 256, 0, stream>>>(src, dst, norm, tmp, agg, E);
  // 9) h1 = relu(agg + c1_b)  (in place)
  bias_relu_kernel<<<cdiv((long long)N * 32, 256), 256, 0, stream>>>(agg, c1_b, N * 32, 32);

  // ---- GCNConv 2 ----
  // 10+11) tmp16 = h1 @ c2_w ; fused: out = dinv^2 * tmp16 + c2_b
  wmma_gemm_kernel<32, 16, false, false, true, true>
      <<<gemmBlocks, 256, 0, stream>>>(agg, c2_w, nullptr, tmp, dinv, out, c2_b, N);
  if (tail)
    gemm_tail_kernel<32, 16, false, false, true, true>
        <<<cdiv((long long)tail * 16, 256), 256, 0, stream>>>(
            agg, c2_w, nullptr, tmp, dinv, out, c2_b, fullWaves * 16, N);
  // 12) out[dst] += norm * tmp16[src]
  scatter_kernel<16><<<cdiv((long long)E * 4, 256), 256, 0, stream>>>(src, dst, norm, tmp, out, E);
}